// LightGCN_45715631898773
// MI455X (gfx1250) — compile-verified
//
#include <hip/hip_runtime.h>

#define N_USERS 70000
#define N_ITEMS 30000
#define NTOT    (N_USERS + N_ITEMS)   // 100000
#define DIM     128
#define NNZ     1000000
#define N_LAYERS 3
#define BATCH   8192

typedef __attribute__((ext_vector_type(2))) float v2f;
typedef __attribute__((ext_vector_type(8))) float v8f;

// ---------------------------------------------------------------------------
// Init: E = concat(user_emb, item_emb); S = E; w_he = deg; w_ho = 1 - deg
// One thread per float4 (NTOT*DIM/4 = 3.2M threads).
// ---------------------------------------------------------------------------
__global__ void k_init(const float* __restrict__ ue, const float* __restrict__ ie,
                       const float* __restrict__ deg,
                       float* __restrict__ E, float* __restrict__ S,
                       float* __restrict__ who, float* __restrict__ whe) {
    int i = blockIdx.x * blockDim.x + threadIdx.x;            // float4 index
    const int user4 = N_USERS * (DIM / 4);
    const int tot4  = NTOT * (DIM / 4);
    if (i < tot4) {
        float4 v = (i < user4) ? ((const float4*)ue)[i]
                               : ((const float4*)ie)[i - user4];
        ((float4*)E)[i] = v;
        ((float4*)S)[i] = v;
    }
    if (i < NTOT) {
        float d = deg[i];
        whe[i] = d;
        who[i] = 1.0f - d;
    }
}

// ---------------------------------------------------------------------------
// SpMM scatter: Y[rows[e]] += vals[e] * X[cols[e]]. One wave32 per edge;
// each lane owns a float4 slice (32 lanes * 4 = 128) -> global_atomic_add_f32.
// ---------------------------------------------------------------------------
__global__ void k_spmm(const int* __restrict__ rows, const int* __restrict__ cols,
                       const float* __restrict__ vals,
                       const float* __restrict__ X, float* __restrict__ Y) {
    int gid  = blockIdx.x * blockDim.x + threadIdx.x;
    int e    = gid >> 5;
    int lane = gid & 31;
    if (e >= NNZ) return;
    int   r = rows[e];
    int   c = cols[e];
    float v = vals[e];
    float4 x = ((const float4*)(X + (size_t)c * DIM))[lane];
    float* y = Y + (size_t)r * DIM + lane * 4;
    __hip_atomic_fetch_add(y + 0, v * x.x, __ATOMIC_RELAXED, __HIP_MEMORY_SCOPE_AGENT);
    __hip_atomic_fetch_add(y + 1, v * x.y, __ATOMIC_RELAXED, __HIP_MEMORY_SCOPE_AGENT);
    __hip_atomic_fetch_add(y + 2, v * x.z, __ATOMIC_RELAXED, __HIP_MEMORY_SCOPE_AGENT);
    __hip_atomic_fetch_add(y + 3, v * x.w, __ATOMIC_RELAXED, __HIP_MEMORY_SCOPE_AGENT);
}

// ---------------------------------------------------------------------------
// Fused per-row layer epilogue (one wave32 per row):
//   new  = w_ho*side_ho + w_he*side_he
//   dho  = dot(new, side_ho); dhe = dot(new, side_he)   (wave reduction)
//   w_ho' = w_ho + 0.1*dho; w_he' = w_he + 0.1*dhe; renormalize
//   E = new; S += new; zero side buffers for the next layer's scatter.
// ---------------------------------------------------------------------------
__global__ void k_combine(float* __restrict__ Pho, float* __restrict__ Phe,
                          float* __restrict__ E, float* __restrict__ S,
                          float* __restrict__ who, float* __restrict__ whe) {
    int gid  = blockIdx.x * blockDim.x + threadIdx.x;
    int r    = gid >> 5;
    int lane = gid & 31;
    if (r >= NTOT) return;

    float wo = who[r];
    float wh = whe[r];
    size_t base = (size_t)r * DIM;
    float4 ho = ((const float4*)(Pho + base))[lane];
    float4 he = ((const float4*)(Phe + base))[lane];

    float4 ne;
    ne.x = wo * ho.x + wh * he.x;
    ne.y = wo * ho.y + wh * he.y;
    ne.z = wo * ho.z + wh * he.z;
    ne.w = wo * ho.w + wh * he.w;

    float pdho = ne.x * ho.x + ne.y * ho.y + ne.z * ho.z + ne.w * ho.w;
    float pdhe = ne.x * he.x + ne.y * he.y + ne.z * he.z + ne.w * he.w;
    #pragma unroll
    for (int off = 16; off > 0; off >>= 1) {      // wave32 butterfly reduce
        pdho += __shfl_xor(pdho, off, 32);
        pdhe += __shfl_xor(pdhe, off, 32);
    }

    ((float4*)(E + base))[lane] = ne;
    float4 s = ((const float4*)(S + base))[lane];
    s.x += ne.x; s.y += ne.y; s.z += ne.z; s.w += ne.w;
    ((float4*)(S + base))[lane] = s;

    // pre-zero the scatter targets for the next layer
    float4 z = make_float4(0.f, 0.f, 0.f, 0.f);
    ((float4*)(Pho + base))[lane] = z;
    ((float4*)(Phe + base))[lane] = z;

    if (lane == 0) {
        float wo2 = wo + 0.1f * pdho;
        float wh2 = wh + 0.1f * pdhe;
        float won = wo2 / (wo2 + wh2);
        who[r] = won;
        whe[r] = 1.0f - won;
    }
}

// ---------------------------------------------------------------------------
// Final batched dot via fp32 WMMA. One wave32 per 16 (user,item) pairs.
// A[m][k] = S[users[tile*16+m]][k], B[k][n] = S[NU+items[tile*16+n]][k];
// D = A x B, gamma = diag(D) / 16 (folds the /4 layer-mean on both operands).
// f32 A/B layout: lane L holds k = k0 + (L<16 ? {0,1} : {2,3}) of row L&15,
// i.e. one contiguous float2 load per lane per WMMA step.
// Launch is an exact wave multiple: EXEC is all-ones as WMMA requires.
// ---------------------------------------------------------------------------
__global__ void k_gamma(const float* __restrict__ S,
                        const int* __restrict__ users, const int* __restrict__ items,
                        float* __restrict__ out) {
    int gid  = blockIdx.x * blockDim.x + threadIdx.x;
    int tile = gid >> 5;                       // 512 tiles of 16 pairs
    int lane = gid & 31;
    int m    = lane & 15;
    int koff = (lane >> 4) << 1;               // 0 for lanes 0-15, 2 for 16-31
    int p    = tile * 16 + m;

    const float* au = S + (size_t)users[p] * DIM + koff;
    const float* bi = S + (size_t)(N_USERS + items[p]) * DIM + koff;

    v8f c = {0.f, 0.f, 0.f, 0.f, 0.f, 0.f, 0.f, 0.f};
    #pragma unroll
    for (int k = 0; k < DIM; k += 4) {
        v2f a = *(const v2f*)(au + k);
        v2f b = *(const v2f*)(bi + k);
        c = __builtin_amdgcn_wmma_f32_16x16x4_f32(false, a, false, b,
                                                  (short)0, c, false, false);
    }

    // diag(D): i<8 -> (vgpr i, lane i); i>=8 -> (vgpr i-8, lane i+16)
    #pragma unroll
    for (int e = 0; e < 8; ++e) {
        float v = c[e] * 0.0625f;              // 1/16 = (1/4)*(1/4)
        if (lane == e)      out[tile * 16 + e]     = v;
        if (lane == 24 + e) out[tile * 16 + 8 + e] = v;
    }
}

// ---------------------------------------------------------------------------
extern "C" void kernel_launch(void* const* d_in, const int* in_sizes, int n_in,
                              void* d_out, int out_size, void* d_ws, size_t ws_size,
                              hipStream_t stream) {
    const float* user_emb = (const float*)d_in[0];
    const float* item_emb = (const float*)d_in[1];
    const int*   he_rows  = (const int*)  d_in[2];
    const int*   he_cols  = (const int*)  d_in[3];
    const float* he_vals  = (const float*)d_in[4];
    const int*   ho_rows  = (const int*)  d_in[5];
    const int*   ho_cols  = (const int*)  d_in[6];
    const float* ho_vals  = (const float*)d_in[7];
    const float* degree   = (const float*)d_in[8];
    const int*   users    = (const int*)  d_in[9];
    const int*   items    = (const int*)  d_in[10];
    float*       out      = (float*)d_out;

    const size_t embBytes = (size_t)NTOT * DIM * sizeof(float);   // 51.2 MB
    const size_t wBytes   = ((size_t)NTOT * sizeof(float) + 511) & ~(size_t)511;
    char* ws = (char*)d_ws;
    float* E   = (float*)(ws);                  // current layer embedding
    float* S   = (float*)(ws + embBytes);       // running sum of layers
    float* Pho = (float*)(ws + 2 * embBytes);   // side_ho scatter buffer
    float* Phe = (float*)(ws + 3 * embBytes);   // side_he scatter buffer
    float* who = (float*)(ws + 4 * embBytes);
    float* whe = (float*)(ws + 4 * embBytes + wBytes);

    // init E, S, weights
    {
        int threads = NTOT * DIM / 4;           // 3,200,000
        k_init<<<(threads + 255) / 256, 256, 0, stream>>>(
            user_emb, item_emb, degree, E, S, who, whe);
    }

    // zero scatter buffers once; k_combine re-zeroes them each layer
    hipMemsetAsync(Pho, 0, embBytes, stream);
    hipMemsetAsync(Phe, 0, embBytes, stream);

    const int spmmBlocks = (NNZ * 32) / 256;    // 125,000
    const int rowBlocks  = (NTOT * 32 + 255) / 256;

    for (int layer = 0; layer < N_LAYERS; ++layer) {
        k_spmm<<<spmmBlocks, 256, 0, stream>>>(ho_rows, ho_cols, ho_vals, E, Pho);
        k_spmm<<<spmmBlocks, 256, 0, stream>>>(he_rows, he_cols, he_vals, E, Phe);
        k_combine<<<rowBlocks, 256, 0, stream>>>(Pho, Phe, E, S, who, whe);
    }

    // gamma: 512 tiles * 32 lanes = 16384 threads = 64 full blocks
    k_gamma<<<(BATCH / 16) * 32 / 256, 256, 0, stream>>>(S, users, items, out);
}